// Cross_modal_TripletLoss_36618891166030
// MI455X (gfx1250) — compile-verified
//
#include <hip/hip_runtime.h>
#include <hip/hip_bf16.h>

typedef __attribute__((ext_vector_type(2))) float v2f;
typedef __attribute__((ext_vector_type(8))) float v8f;
typedef unsigned int u32;
typedef __attribute__((ext_vector_type(4))) u32 u32x4;
typedef __attribute__((ext_vector_type(8))) u32 u32x8;

#define TILE_M 128
#define TILE_N 64
#define KC 32
#define LDS_STRIDE 36   // 32 data dwords + 4 pad dwords (TDM pad_interval=32dw, pad_amount=4dw)
#define NTHREADS 256

// ---------------------------------------------------------------------------
// Kernel 1: row squared-norms for both modalities + init of ap/an bit arrays.
// ---------------------------------------------------------------------------
__global__ __launch_bounds__(256) void norms_init_kernel(
    const float* __restrict__ m1, const float* __restrict__ m2, int n, int d,
    float* __restrict__ sq1, float* __restrict__ sq2,
    unsigned* __restrict__ apb, unsigned* __restrict__ anb)
{
    const int tid = threadIdx.x;
    const long long gid = (long long)blockIdx.x * blockDim.x + tid;
    if (gid < n) { apb[gid] = 0u; anb[gid] = 0x7F800000u; }

    const int wave = (int)(gid >> 5);
    const int lane = tid & 31;
    if (wave < 2 * n) {
        const float* src = (wave < n) ? (m1 + (size_t)wave * d)
                                      : (m2 + (size_t)(wave - n) * d);
        float s = 0.f;
        for (int i = lane; i < d; i += 32) { float v = src[i]; s += v * v; }
        #pragma unroll
        for (int off = 16; off > 0; off >>= 1) s += __shfl_xor(s, off, 32);
        if (lane == 0) {
            if (wave < n) sq1[wave] = s; else sq2[wave - n] = s;
        }
    }
}

// ---------------------------------------------------------------------------
// Tensor Data Mover: DMA a (rows x KC) fp32 tile (row stride d elements) into
// LDS with hardware padding -> 36-dword LDS row stride. Descriptor bitfields
// per CDNA5 ISA ch.8 (D# group0 / group1). Wave-level op; EXEC ignored.
// ---------------------------------------------------------------------------
__device__ __forceinline__ void tdm_load_tile(const float* gptr, u32 lds_byte,
                                              int tile_rows, int n_rows, int d_elems)
{
    const unsigned long long ga = (unsigned long long)(uintptr_t)gptr;
    u32x4 g0;
    g0.x = 1u;                                          // count=1, user mode, no gather
    g0.y = lds_byte;                                    // lds_addr (bytes)
    g0.z = (u32)ga;                                     // global_addr[31:0]
    g0.w = (u32)((ga >> 32) & 0x1FFFFFFu) | (2u << 30); // global_addr[56:32] | type=2

    const u32 dim0 = (u32)d_elems;     // tensor dim0 (elements along K)
    const u32 dim1 = (u32)n_rows;      // tensor dim1 (rows)
    u32x8 g1;
    g1.s0 = (2u << 16)                 // data_size: 4 bytes
          | (1u << 20)                 // pad_enable
          | (4u << 22)                 // pad_interval: every 32 dwords
          | (3u << 25);                // pad_amount: 4 dwords -> 36-dword row stride
    g1.s1 = dim0 << 16;                                 // tensor_dim0[15:0]
    g1.s2 = (dim0 >> 16) | (dim1 << 16);                // tensor_dim0[31:16] | tensor_dim1[15:0]
    g1.s3 = (dim1 >> 16) | ((u32)KC << 16);             // tensor_dim1[31:16] | tile_dim0
    g1.s4 = (u32)tile_rows;                             // tile_dim1 (tile_dim2=0)
    g1.s5 = (u32)d_elems;                               // tensor_dim0_stride[31:0]
    g1.s6 = 0u;                                         // stride0[47:32]=0 | stride1[15:0]=0
    g1.s7 = 0u;                                         // stride1[47:16]=0

    asm volatile("tensor_load_to_lds %0, %1" :: "s"(g0), "s"(g1) : "memory");
}

// ---------------------------------------------------------------------------
// Kernel 2: fused fp32 WMMA GEMM (M1 * M2^T) + distance + masked row max/min.
// WG = 8 waves; tile = 128 rows x 64 cols; K staged by TDM, double-buffered.
// ---------------------------------------------------------------------------
__global__ __launch_bounds__(256) void triplet_gemm_kernel(
    const float* __restrict__ m1, const float* __restrict__ m2,
    const long long* __restrict__ targets,
    const float* __restrict__ sq1, const float* __restrict__ sq2,
    unsigned* __restrict__ apb, unsigned* __restrict__ anb,
    int n, int d, int col_chunks)
{
    __shared__ float As[2][TILE_M * LDS_STRIDE];   // 2 x 18 KB
    __shared__ float Bs[2][TILE_N * LDS_STRIDE];   // 2 x  9 KB
    __shared__ float s_sq1[TILE_M];
    __shared__ float s_sq2[TILE_N];
    __shared__ int s_tr[TILE_M];
    __shared__ int s_tc[TILE_N];
    __shared__ unsigned s_ap[TILE_M];
    __shared__ unsigned s_an[TILE_M];

    const int tid = threadIdx.x;
    const int lane = tid & 31;
    const int wv = tid >> 5;
    const int row0 = blockIdx.x * TILE_M;

    const u32 ldsA0 = (u32)(uintptr_t)&As[0][0];
    const u32 ldsA1 = (u32)(uintptr_t)&As[1][0];
    const u32 ldsB0 = (u32)(uintptr_t)&Bs[0][0];
    const u32 ldsB1 = (u32)(uintptr_t)&Bs[1][0];

    if (tid < TILE_M) {
        s_ap[tid] = 0u;
        s_an[tid] = 0x7F800000u;
        s_sq1[tid] = sq1[row0 + tid];
        s_tr[tid] = (int)targets[row0 + tid];
    }

    for (int cc = 0; cc < col_chunks; ++cc) {
        const int col0 = (blockIdx.y * col_chunks + cc) * TILE_N;
        __syncthreads();   // previous epilogue done before s_sq2/s_tc rewrite / buffer reuse
        if (tid < TILE_N) {
            s_sq2[tid] = sq2[col0 + tid];
            s_tc[tid] = (int)targets[col0 + tid];
        }

        v8f acc[4] = {};

        // Prologue: wave 0 launches TDM for the first K chunk into buffer 0.
        if (wv == 0) {
            tdm_load_tile(m1 + (size_t)row0 * d, ldsA0, TILE_M, n, d);
            tdm_load_tile(m2 + (size_t)col0 * d, ldsB0, TILE_N, n, d);
        }

        int buf = 0;
        for (int kc = 0; kc < d; kc += KC) {
            if (wv == 0) __builtin_amdgcn_s_wait_tensorcnt(0);  // tile[buf] landed
            __syncthreads();                                    // publish to all waves
            // Pipeline: DMA next chunk into the other buffer while we compute.
            if (wv == 0 && kc + KC < d) {
                tdm_load_tile(m1 + (size_t)row0 * d + (kc + KC),
                              buf ? ldsA0 : ldsA1, TILE_M, n, d);
                tdm_load_tile(m2 + (size_t)col0 * d + (kc + KC),
                              buf ? ldsB0 : ldsB1, TILE_N, n, d);
            }

            const float* __restrict__ a_tile = &As[buf][0];
            const float* __restrict__ b_tile = &Bs[buf][0];
            const int khalf = (lane >> 4) << 1;   // lanes 16-31 hold K+2,K+3
            const int rsel = lane & 15;
            #pragma unroll
            for (int k = 0; k < KC; k += 4) {
                const v2f a = *(const v2f*)&a_tile[(wv * 16 + rsel) * LDS_STRIDE + k + khalf];
                #pragma unroll
                for (int s = 0; s < 4; ++s) {
                    const v2f b = *(const v2f*)&b_tile[(s * 16 + rsel) * LDS_STRIDE + k + khalf];
                    acc[s] = __builtin_amdgcn_wmma_f32_16x16x4_f32(
                        false, a, false, b, (short)0, acc[s], false, false);
                }
            }
            buf ^= 1;
        }

        // Epilogue: dist = sqrt(clip(sq1 + sq2 - 2*dot, eps)); masked max/min
        // via LDS u32 atomics (bit order of positive floats == float order).
        const int rhalf = (lane >> 4) << 3;
        const int nsel = lane & 15;
        #pragma unroll
        for (int s = 0; s < 4; ++s) {
            #pragma unroll
            for (int r = 0; r < 8; ++r) {
                const int rl = wv * 16 + r + rhalf;
                const int cl = s * 16 + nsel;
                const float g = acc[s][r];
                const float dd = s_sq1[rl] + s_sq2[cl] - 2.0f * g;
                const float dist = sqrtf(fmaxf(dd, 1e-12f));
                const unsigned bits = __float_as_uint(dist);
                if (s_tr[rl] == s_tc[cl]) atomicMax(&s_ap[rl], bits);
                else                      atomicMin(&s_an[rl], bits);
            }
        }
    }

    __syncthreads();
    if (tid < TILE_M) {
        atomicMax(&apb[row0 + tid], s_ap[tid]);
        atomicMin(&anb[row0 + tid], s_an[tid]);
    }
}

// ---------------------------------------------------------------------------
// Kernel 3: final scalar reduction -> (loss, prec).
// ---------------------------------------------------------------------------
__global__ __launch_bounds__(256) void finalize_kernel(
    const unsigned* __restrict__ apb, const unsigned* __restrict__ anb,
    const float* __restrict__ margin, int n, float* __restrict__ out)
{
    __shared__ float s_l[256];
    __shared__ float s_p[256];
    const int tid = threadIdx.x;
    const float m = margin[0];
    float ls = 0.f, ps = 0.f;
    for (int i = tid; i < n; i += 256) {
        const float ap = __uint_as_float(apb[i]);
        const float an = __uint_as_float(anb[i]);
        ls += fmaxf(ap - an + m, 0.f);
        ps += (an > ap) ? 1.f : 0.f;
    }
    s_l[tid] = ls; s_p[tid] = ps;
    __syncthreads();
    #pragma unroll
    for (int off = 128; off > 0; off >>= 1) {
        if (tid < off) { s_l[tid] += s_l[tid + off]; s_p[tid] += s_p[tid + off]; }
        __syncthreads();
    }
    if (tid == 0) { out[0] = s_l[0] / (float)n; out[1] = s_p[0] / (float)n; }
}

extern "C" void kernel_launch(void* const* d_in, const int* in_sizes, int n_in,
                              void* d_out, int out_size, void* d_ws, size_t ws_size,
                              hipStream_t stream) {
    const float* m1 = (const float*)d_in[0];
    const float* m2 = (const float*)d_in[1];
    const long long* tg = (const long long*)d_in[2];   // int64 targets
    const float* margin = (const float*)d_in[3];
    float* out = (float*)d_out;

    const int n = in_sizes[2];          // 4096
    const int d = in_sizes[0] / n;      // 2048

    float* sq1 = (float*)d_ws;
    float* sq2 = sq1 + n;
    unsigned* apb = (unsigned*)(sq2 + n);
    unsigned* anb = apb + n;

    const int norm_blocks = (2 * n * 32 + 255) / 256;
    norms_init_kernel<<<norm_blocks, 256, 0, stream>>>(m1, m2, n, d, sq1, sq2, apb, anb);

    const int ysplit = 8;
    const int col_chunks = n / (ysplit * TILE_N);       // 8
    dim3 grid(n / TILE_M, ysplit);                      // (32, 8)
    triplet_gemm_kernel<<<grid, NTHREADS, 0, stream>>>(m1, m2, tg, sq1, sq2,
                                                       apb, anb, n, d, col_chunks);

    finalize_kernel<<<1, 256, 0, stream>>>(apb, anb, margin, n, out);
}